// DeepMDsimpleEnergyNUFFT_84361747628285
// MI455X (gfx1250) — compile-verified
//
#include <hip/hip_runtime.h>
#include <hip/hip_bf16.h>

// ---------------------------------------------------------------------------
// DeepMD-like energy + NUFFT long range, forward + VJP forces, for gfx1250.
// All big contractions are wave32 WMMA f32 16x16x4 GEMMs; the Kronecker
// factor Eyz = Ey*Ez (961 wide) is generated on the fly in registers.
// Inner loops are branchless (pad handling via clamped index + mask mult)
// so EXEC stays all-ones around every v_wmma.
// ---------------------------------------------------------------------------

constexpr int S_  = 8;
constexpr int N_  = 1024;
constexpr int M_  = 64;
constexpr int KM_ = 31;          // Fourier modes per axis
constexpr int C_  = 2;           // channels
constexpr int BCP_ = 976;        // 31*31 = 961 padded to 61*16
constexpr float LBOX = 16.0f;
#define TWOPI_ 6.283185307179586f
constexpr float INVL3 = 1.0f / (16.0f * 16.0f * 16.0f);

typedef float v2f __attribute__((ext_vector_type(2)));
typedef float v8f __attribute__((ext_vector_type(8)));

__device__ inline float kvalf(int i) { return (TWOPI_ / LBOX) * (float)(i - 15); }

__device__ inline v8f zero8() { v8f z = {0.f,0.f,0.f,0.f,0.f,0.f,0.f,0.f}; return z; }

__device__ inline v8f wmma4(v2f a, v2f b, v8f c) {
  // D = A(16x4,f32) * B(4x16,f32) + C(16x16,f32)
  return __builtin_amdgcn_wmma_f32_16x16x4_f32(false, a, false, b, (short)0, c,
                                               false, false);
}

// ----------------------------- tiny MLP helpers ----------------------------

template <int DI, int DN>
__device__ inline void fwd_tanh(const float* x, const float* W, const float* b,
                                float* y) {
#pragma unroll
  for (int j = 0; j < DN; j++) {
    float acc = b[j];
#pragma unroll
    for (int i = 0; i < DI; i++) acc += x[i] * W[i * DN + j];
    y[j] = tanhf(acc);
  }
}

template <int DI, int DN>
__device__ inline void fwd_relu(const float* x, const float* W, const float* b,
                                float* y) {
#pragma unroll
  for (int j = 0; j < DN; j++) {
    float acc = b[j];
#pragma unroll
    for (int i = 0; i < DI; i++) acc += x[i] * W[i * DN + j];
    y[j] = fmaxf(acc, 0.0f);
  }
}

template <int DI, int DN>
__device__ inline void fwd_tanh_dual(const float* x, const float* dx,
                                     const float* W, const float* b, float* y,
                                     float* dy) {
#pragma unroll
  for (int j = 0; j < DN; j++) {
    float acc = b[j], dacc = 0.f;
#pragma unroll
    for (int i = 0; i < DI; i++) {
      acc += x[i] * W[i * DN + j];
      dacc += dx[i] * W[i * DN + j];
    }
    float tv = tanhf(acc);
    y[j] = tv;
    dy[j] = (1.f - tv * tv) * dacc;
  }
}

template <int DI, int DN>
__device__ inline void bwd_tanh(const float* y, const float* W, const float* gy,
                                float* gx) {
  float gp[DN];
#pragma unroll
  for (int j = 0; j < DN; j++) gp[j] = gy[j] * (1.f - y[j] * y[j]);
#pragma unroll
  for (int i = 0; i < DI; i++) {
    float a = 0.f;
#pragma unroll
    for (int j = 0; j < DN; j++) a += W[i * DN + j] * gp[j];
    gx[i] = a;
  }
}

template <int DI, int DN>
__device__ inline void bwd_relu(const float* y, const float* W, const float* gy,
                                float* gx) {
  float gp[DN];
#pragma unroll
  for (int j = 0; j < DN; j++) gp[j] = (y[j] > 0.f) ? gy[j] : 0.f;
#pragma unroll
  for (int i = 0; i < DI; i++) {
    float a = 0.f;
#pragma unroll
    for (int j = 0; j < DN; j++) a += W[i * DN + j] * gp[j];
    gx[i] = a;
  }
}

// ------------------------------- kernels -----------------------------------

__global__ void k_zero(float* p, int n) {
  int i = blockIdx.x * blockDim.x + threadIdx.x;
  if (i < n) p[i] = 0.f;
}

// Per (s,n): Ex,Ey,Ez = exp(-i * k * coord), stored [s][k][n].
__global__ void k_phase(const float* __restrict__ inp, float* exr, float* exi,
                        float* eyr, float* eyi, float* ezr, float* ezi) {
  int t = blockIdx.x * blockDim.x + threadIdx.x;
  if (t >= S_ * N_) return;
  int s = t / N_, n = t % N_;
  float x = inp[(size_t)t * 3 + 0];
  float y = inp[(size_t)t * 3 + 1];
  float z = inp[(size_t)t * 3 + 2];
  size_t base = (size_t)s * KM_ * N_ + n;
  for (int k = 0; k < KM_; k++) {
    float kv = kvalf(k);
    float sx, cx, sy, cy, sz, cz;
    __sincosf(kv * x, &sx, &cx);
    __sincosf(kv * y, &sy, &cy);
    __sincosf(kv * z, &sz, &cz);
    size_t o = base + (size_t)k * N_;
    exr[o] = cx; exi[o] = -sx;
    eyr[o] = cy; eyi[o] = -sy;
    ezr[o] = cz; ezi[o] = -sz;
  }
}

// Descriptor forward: one block (64 threads) per (s,n); thread = neighbor m.
__global__ __launch_bounds__(64) void k_desc_fwd(
    const float* __restrict__ inp, const int* __restrict__ neigh,
    const float* W0, const float* W1, const float* W2, const float* W3,
    const float* W4, const float* b0, const float* b1, const float* b2,
    const float* b3, const float* b4, float* __restrict__ D) {
  int sn = blockIdx.x;
  int s = sn / N_;
  int m = threadIdx.x;
  __shared__ float Dsh[32];
  if (m < 32) Dsh[m] = 0.f;
  __syncthreads();
  int j = neigh[(size_t)sn * M_ + m];
  const float* pi = inp + (size_t)sn * 3;
  const float* pj = inp + ((size_t)s * N_ + j) * 3;
  float dx = pj[0] - pi[0]; dx -= LBOX * roundf(dx * (1.0f / LBOX));
  float dy = pj[1] - pi[1]; dy -= LBOX * roundf(dy * (1.0f / LBOX));
  float dz = pj[2] - pi[2]; dz -= LBOX * roundf(dz * (1.0f / LBOX));
  float r2 = dx * dx + dy * dy + dz * dz;
  float sv = 1.0f / sqrtf(r2);
  float x0[1] = {sv};
  float h0[2], h1[4], h2[8], h3[16], L1[32];
  fwd_tanh<1, 2>(x0, W0, b0, h0);
  fwd_tanh<2, 4>(h0, W1, b1, h1);
  fwd_tanh<4, 8>(h1, W2, b2, h2);
  fwd_tanh<8, 16>(h2, W3, b3, h3);
  fwd_tanh<16, 32>(h3, W4, b4, L1);
#pragma unroll
  for (int k = 0; k < 32; k++) atomicAdd(&Dsh[k], L1[k] * sv);
  __syncthreads();
  if (m < 32) D[(size_t)sn * 32 + m] = Dsh[m];
}

// Descriptor backward: recompute with forward-mode dual, scatter forces.
__global__ __launch_bounds__(64) void k_desc_bwd(
    const float* __restrict__ inp, const int* __restrict__ neigh,
    const float* W0, const float* W1, const float* W2, const float* W3,
    const float* W4, const float* b0, const float* b1, const float* b2,
    const float* b3, const float* b4, const float* __restrict__ gD,
    float* __restrict__ Fout) {
  int sn = blockIdx.x;
  int s = sn / N_;
  int m = threadIdx.x;
  int j = neigh[(size_t)sn * M_ + m];
  const float* pi = inp + (size_t)sn * 3;
  const float* pj = inp + ((size_t)s * N_ + j) * 3;
  float dx = pj[0] - pi[0]; dx -= LBOX * roundf(dx * (1.0f / LBOX));
  float dy = pj[1] - pi[1]; dy -= LBOX * roundf(dy * (1.0f / LBOX));
  float dz = pj[2] - pi[2]; dz -= LBOX * roundf(dz * (1.0f / LBOX));
  float r2 = dx * dx + dy * dy + dz * dz;
  float rin = 1.0f / sqrtf(r2);
  float sv = rin;
  float x0[1] = {sv}, dx0[1] = {1.0f};
  float h0[2], d0[2], h1[4], d1[4], h2[8], d2[8], h3[16], d3[16];
  float L1[32], dL1[32];
  fwd_tanh_dual<1, 2>(x0, dx0, W0, b0, h0, d0);
  fwd_tanh_dual<2, 4>(h0, d0, W1, b1, h1, d1);
  fwd_tanh_dual<4, 8>(h1, d1, W2, b2, h2, d2);
  fwd_tanh_dual<8, 16>(h2, d2, W3, b3, h3, d3);
  fwd_tanh_dual<16, 32>(h3, d3, W4, b4, L1, dL1);
  float gs = 0.f;
  const float* g = gD + (size_t)sn * 32;
#pragma unroll
  for (int k = 0; k < 32; k++) gs += g[k] * (L1[k] + sv * dL1[k]);
  // dE/dd = gs * ds/dr * d/r = -gs * d / r^3
  float c = -gs * rin * rin * rin;
  float gdx = c * dx, gdy = c * dy, gdz = c * dz;
  // F = -grad: F_i = +dE/dd, F_j = -dE/dd
  atomicAdd(&Fout[(size_t)sn * 3 + 0], gdx);
  atomicAdd(&Fout[(size_t)sn * 3 + 1], gdy);
  atomicAdd(&Fout[(size_t)sn * 3 + 2], gdz);
  size_t jb = ((size_t)s * N_ + j) * 3;
  atomicAdd(&Fout[jb + 0], -gdx);
  atomicAdd(&Fout[jb + 1], -gdy);
  atomicAdd(&Fout[jb + 2], -gdz);
}

// C[a(32 pad), bc(976 pad)] = sum_n (w? * Ex^sA)[a,n] * (Ey*Ez)^sB[n,bc]
// complex; one wave per block; WMMA f32 16x16x4, K = n.
// Branchless: pad lanes use clamped addresses and a 0/1 mask multiply, so
// EXEC stays all-ones through the whole K loop (required for WMMA anyway).
template <bool HASWT>
__global__ __launch_bounds__(32) void k_gemm_fhat(
    const float* __restrict__ exr, const float* __restrict__ exi,
    const float* __restrict__ eyr, const float* __restrict__ eyi,
    const float* __restrict__ ezr, const float* __restrict__ ezi,
    const float* __restrict__ wt, int nch, float sA, float sB,
    float* __restrict__ outR, float* __restrict__ outI) {
  const int bct = blockIdx.x, at = blockIdx.y, sc = blockIdx.z;
  const int s = sc / nch, ch = sc % nch;
  const int lane = threadIdx.x;
  const int l15 = lane & 15;
  const int kh = (lane >> 4) & 1;
  const int arow = at * 16 + l15;
  const int bcol = bct * 16 + l15;
  const float amask = (arow < KM_) ? 1.f : 0.f;
  const float bmask = (bcol < KM_ * KM_) ? 1.f : 0.f;
  const int arc = (arow < KM_) ? arow : (KM_ - 1);
  const int bcc = (bcol < KM_ * KM_) ? bcol : (KM_ * KM_ - 1);
  const int b_ = bcc / KM_;
  const int c_ = bcc % KM_;
  const float aim = sA * amask;  // fold conj sign + pad mask
  const float* Exr = exr + (size_t)s * KM_ * N_ + (size_t)arc * N_;
  const float* Exi = exi + (size_t)s * KM_ * N_ + (size_t)arc * N_;
  const float* Eyr = eyr + (size_t)s * KM_ * N_ + (size_t)b_ * N_;
  const float* Eyi = eyi + (size_t)s * KM_ * N_ + (size_t)b_ * N_;
  const float* Ezr = ezr + (size_t)s * KM_ * N_ + (size_t)c_ * N_;
  const float* Ezi = ezi + (size_t)s * KM_ * N_ + (size_t)c_ * N_;
  const float* Wp = HASWT ? (wt + (size_t)s * N_ * C_ + ch) : nullptr;
  v8f aRR = zero8(), aRI = zero8(), aIR = zero8(), aII = zero8();
  for (int n0 = 0; n0 < N_; n0 += 4) {
    const int n = n0 + kh * 2;
    float w0 = 1.f, w1 = 1.f;
    if (HASWT) {
      w0 = Wp[(size_t)n * C_];
      w1 = Wp[(size_t)(n + 1) * C_];
    }
    v2f ar, ai, br, bi;
    ar.x = Exr[n] * (w0 * amask);     ar.y = Exr[n + 1] * (w1 * amask);
    ai.x = Exi[n] * (w0 * aim);       ai.y = Exi[n + 1] * (w1 * aim);
    {
      float yr0 = Eyr[n], yi0 = sB * Eyi[n];
      float zr0 = Ezr[n], zi0 = sB * Ezi[n];
      float yr1 = Eyr[n + 1], yi1 = sB * Eyi[n + 1];
      float zr1 = Ezr[n + 1], zi1 = sB * Ezi[n + 1];
      br.x = (yr0 * zr0 - yi0 * zi0) * bmask;
      bi.x = (yr0 * zi0 + yi0 * zr0) * bmask;
      br.y = (yr1 * zr1 - yi1 * zi1) * bmask;
      bi.y = (yr1 * zi1 + yi1 * zr1) * bmask;
    }
    aRR = wmma4(ar, br, aRR);
    aRI = wmma4(ar, bi, aRI);
    aIR = wmma4(ai, br, aIR);
    aII = wmma4(ai, bi, aII);
  }
  const int ccol = bct * 16 + l15;
  const int rb = at * 16 + kh * 8;
  float* OR = outR + (size_t)sc * 32 * BCP_;
  float* OI = outI + (size_t)sc * 32 * BCP_;
#pragma unroll
  for (int v = 0; v < 8; v++) {
    int r = rb + v;
    OR[(size_t)r * BCP_ + ccol] = aRR[v] - aII[v];
    OI[(size_t)r * BCP_ + ccol] = aRI[v] + aIR[v];
  }
}

// G[a(32), n(1024)] = sum_bc H[a,bc] * ((Ey*Ez)^sB * kw)[bc,n]  (complex)
// KMODE: 0 none, 1 weight k_b, 2 weight k_c.  K = bc (padded, pads read 0
// from H; B pads masked to 0 via clamped index + mask multiply).
template <int KMODE>
__global__ __launch_bounds__(32) void k_gemm_g(
    const float* __restrict__ HRm, const float* __restrict__ HIm,
    const float* __restrict__ eyr, const float* __restrict__ eyi,
    const float* __restrict__ ezr, const float* __restrict__ ezi, int nch,
    float sB, float* __restrict__ GRo, float* __restrict__ GIo) {
  const int nt = blockIdx.x, at = blockIdx.y, sc = blockIdx.z;
  const int s = sc / nch;
  const int lane = threadIdx.x;
  const int l15 = lane & 15;
  const int kh = (lane >> 4) & 1;
  const int arow = at * 16 + l15;
  const int ncol = nt * 16 + l15;
  const float* HR = HRm + (size_t)sc * 32 * BCP_ + (size_t)arow * BCP_;
  const float* HI = HIm + (size_t)sc * 32 * BCP_ + (size_t)arow * BCP_;
  const float* Eyr = eyr + (size_t)s * KM_ * N_ + ncol;
  const float* Eyi = eyi + (size_t)s * KM_ * N_ + ncol;
  const float* Ezr = ezr + (size_t)s * KM_ * N_ + ncol;
  const float* Ezi = ezi + (size_t)s * KM_ * N_ + ncol;
  v8f aRR = zero8(), aRI = zero8(), aIR = zero8(), aII = zero8();
  for (int k0 = 0; k0 < BCP_; k0 += 4) {
    const int kk = k0 + kh * 2;
    v2f ar, ai, br, bi;
    ar.x = HR[kk];  ar.y = HR[kk + 1];
    ai.x = HI[kk];  ai.y = HI[kk + 1];
#pragma unroll
    for (int jj = 0; jj < 2; jj++) {
      const int bc = kk + jj;
      const float msk = (bc < KM_ * KM_) ? 1.f : 0.f;
      const int bcc = (bc < KM_ * KM_) ? bc : (KM_ * KM_ - 1);
      const int b_ = bcc / KM_, c_ = bcc % KM_;
      const float yr = Eyr[(size_t)b_ * N_], yi = sB * Eyi[(size_t)b_ * N_];
      const float zr = Ezr[(size_t)c_ * N_], zi = sB * Ezi[(size_t)c_ * N_];
      float w = msk;
      if (KMODE == 1) w = kvalf(b_) * msk;
      if (KMODE == 2) w = kvalf(c_) * msk;
      const float pr = (yr * zr - yi * zi) * w;
      const float pi2 = (yr * zi + yi * zr) * w;
      if (jj == 0) { br.x = pr; bi.x = pi2; }
      else         { br.y = pr; bi.y = pi2; }
    }
    aRR = wmma4(ar, br, aRR);
    aRI = wmma4(ar, bi, aRI);
    aIR = wmma4(ai, br, aIR);
    aII = wmma4(ai, bi, aII);
  }
  const int rb = at * 16 + kh * 8;
  float* GR = GRo + (size_t)sc * 32 * N_;
  float* GI = GIo + (size_t)sc * 32 * N_;
#pragma unroll
  for (int v = 0; v < 8; v++) {
    int r = rb + v;
    GR[(size_t)r * N_ + ncol] = aRR[v] - aII[v];
    GI[(size_t)r * N_ + ncol] = aRI[v] + aIR[v];
  }
}

// h = mult(ch,k) * f_hat   (pads carry through as zero)
__global__ void k_mult(const float* __restrict__ fR, const float* __restrict__ fI,
                       const float* __restrict__ amp, const float* __restrict__ mu,
                       float* __restrict__ hR, float* __restrict__ hI) {
  size_t t = (size_t)blockIdx.x * blockDim.x + threadIdx.x;
  const size_t total = (size_t)S_ * C_ * 32 * BCP_;
  if (t >= total) return;
  int bc = (int)(t % BCP_);
  int a = (int)((t / BCP_) % 32);
  int ch = (int)((t / ((size_t)BCP_ * 32)) % C_);
  int s = (int)(t / ((size_t)BCP_ * 32 * C_));
  int b = bc / KM_, c = bc % KM_;
  float ka = kvalf(a), kb = kvalf(b), kc = kvalf(c);
  float k2 = ka * ka + kb * kb + kc * kc;
  float mm = mu[ch];
  float mlt = (2.0f * TWOPI_) * amp[ch] / (k2 + mm * mm);
  size_t fidx = ((size_t)s * 32 + a) * BCP_ + bc;
  size_t hidx = ((size_t)(s * C_ + ch) * 32 + a) * BCP_ + bc;
  hR[hidx] = mlt * fR[fidx];
  hI[hidx] = mlt * fI[fidx];
}

// W2 = sum_ch mult(ch) * Wh[ch]
__global__ void k_w2(const float* __restrict__ WhR, const float* __restrict__ WhI,
                     const float* __restrict__ amp, const float* __restrict__ mu,
                     float* __restrict__ W2R, float* __restrict__ W2I) {
  size_t t = (size_t)blockIdx.x * blockDim.x + threadIdx.x;
  const size_t total = (size_t)S_ * 32 * BCP_;
  if (t >= total) return;
  int bc = (int)(t % BCP_);
  int a = (int)((t / BCP_) % 32);
  int s = (int)(t / ((size_t)BCP_ * 32));
  int b = bc / KM_, c = bc % KM_;
  float ka = kvalf(a), kb = kvalf(b), kc = kvalf(c);
  float k2 = ka * ka + kb * kb + kc * kc;
  float accR = 0.f, accI = 0.f;
  for (int ch = 0; ch < C_; ch++) {
    float mm = mu[ch];
    float mlt = (2.0f * TWOPI_) * amp[ch] / (k2 + mm * mm);
    size_t widx = ((size_t)(s * C_ + ch) * 32 + a) * BCP_ + bc;
    accR += mlt * WhR[widx];
    accI += mlt * WhI[widx];
  }
  W2R[t] = accR;
  W2I[t] = accI;
}

// u[s,n,ch] = invL3 * sum_a Re(conj(Ex) * G) = invL3 * sum_a (xr*Gr + xi*Gi)
__global__ void k_u(const float* __restrict__ exr, const float* __restrict__ exi,
                    const float* __restrict__ GR, const float* __restrict__ GI,
                    float* __restrict__ lr) {
  int t = blockIdx.x * blockDim.x + threadIdx.x;
  if (t >= S_ * N_) return;
  int s = t / N_, n = t % N_;
  size_t eb = (size_t)s * KM_ * N_ + n;
  for (int ch = 0; ch < C_; ch++) {
    size_t gb = (size_t)(s * C_ + ch) * 32 * N_ + n;
    float acc = 0.f;
    for (int a = 0; a < KM_; a++) {
      float xr = exr[eb + (size_t)a * N_], xi = exi[eb + (size_t)a * N_];
      acc += xr * GR[gb + (size_t)a * N_] + xi * GI[gb + (size_t)a * N_];
    }
    lr[(size_t)t * C_ + ch] = acc * INVL3;
  }
}

// Per-atom fit + lr MLP forward, energy accumulation, and analytic backward
// producing gD (desc adjoint) and gu (long-range adjoint).
__global__ void k_mlp(const float* __restrict__ Db, const float* __restrict__ ub,
                      const float* lW0, const float* lW1, const float* lW2,
                      const float* lW3, const float* lW4, const float* lb0,
                      const float* lb1, const float* lb2, const float* lb3,
                      const float* lb4, const float* fW0, const float* fW1,
                      const float* fW2, const float* fW3, const float* fW4,
                      const float* fb0, const float* fb1, const float* fb2,
                      const float* fb3, const float* fb4, const float* linW,
                      float* __restrict__ Eout, float* __restrict__ gDb,
                      float* __restrict__ gub) {
  int t = blockIdx.x * blockDim.x + threadIdx.x;
  if (t >= S_ * N_) return;
  int s = t / N_;
  float xin[2] = {ub[(size_t)t * 2 + 0], ub[(size_t)t * 2 + 1]};
  float h0[2], h1[4], h2[8], h3[16], h4[32];
  fwd_relu<2, 2>(xin, lW0, lb0, h0);
  fwd_relu<2, 4>(h0, lW1, lb1, h1);
  fwd_relu<4, 8>(h1, lW2, lb2, h2);
  fwd_relu<8, 16>(h2, lW3, lb3, h3);
  fwd_relu<16, 32>(h3, lW4, lb4, h4);
  float cat[64];
#pragma unroll
  for (int k = 0; k < 32; k++) {
    cat[k] = Db[(size_t)t * 32 + k];
    cat[32 + k] = h4[k];
  }
  float t0[16], t1[8], t2[4], t3[2], t4[1];
  fwd_tanh<64, 16>(cat, fW0, fb0, t0);
  fwd_tanh<16, 8>(t0, fW1, fb1, t1);
  fwd_tanh<8, 4>(t1, fW2, fb2, t2);
  fwd_tanh<4, 2>(t2, fW3, fb3, t3);
  fwd_tanh<2, 1>(t3, fW4, fb4, t4);
  float F = t4[0] * linW[0];
  atomicAdd(&Eout[s], F);
  // backward (dE/dF = 1)
  float g4[1] = {linW[0]};
  float g3[2]; bwd_tanh<2, 1>(t4, fW4, g4, g3);
  float g2[4]; bwd_tanh<4, 2>(t3, fW3, g3, g2);
  float g1[8]; bwd_tanh<8, 4>(t2, fW2, g2, g1);
  float g0[16]; bwd_tanh<16, 8>(t1, fW1, g1, g0);
  float gcat[64]; bwd_tanh<64, 16>(t0, fW0, g0, gcat);
#pragma unroll
  for (int k = 0; k < 32; k++) gDb[(size_t)t * 32 + k] = gcat[k];
  float gh4[32];
#pragma unroll
  for (int k = 0; k < 32; k++) gh4[k] = gcat[32 + k];
  float gh3[16]; bwd_relu<16, 32>(h4, lW4, gh4, gh3);
  float gh2[8];  bwd_relu<8, 16>(h3, lW3, gh3, gh2);
  float gh1[4];  bwd_relu<4, 8>(h2, lW2, gh2, gh1);
  float gh0[2];  bwd_relu<2, 4>(h1, lW1, gh1, gh0);
  float gin[2];  bwd_relu<2, 2>(h0, lW0, gh0, gin);
  gub[(size_t)t * 2 + 0] = gin[0];
  gub[(size_t)t * 2 + 1] = gin[1];
}

// NUFFT force assembly: direct (part A, via G/Gy/Gz) + through-f_hat
// (part B, via GB/GBy/GBz).
__global__ void k_nufft_forces(
    const float* __restrict__ exr, const float* __restrict__ exi,
    const float* __restrict__ GR, const float* __restrict__ GI,
    const float* __restrict__ GyR, const float* __restrict__ GyI,
    const float* __restrict__ GzR, const float* __restrict__ GzI,
    const float* __restrict__ GBR, const float* __restrict__ GBI,
    const float* __restrict__ GByR, const float* __restrict__ GByI,
    const float* __restrict__ GBzR, const float* __restrict__ GBzI,
    const float* __restrict__ gu, float* __restrict__ Fout) {
  int t = blockIdx.x * blockDim.x + threadIdx.x;
  if (t >= S_ * N_) return;
  int s = t / N_, n = t % N_;
  float g0 = gu[(size_t)t * 2 + 0], g1 = gu[(size_t)t * 2 + 1];
  float gx = 0.f, gy = 0.f, gz = 0.f;
  size_t eb = (size_t)s * KM_ * N_ + n;
  size_t gb0 = (size_t)(s * C_ + 0) * 32 * N_ + n;
  size_t gb1 = (size_t)(s * C_ + 1) * 32 * N_ + n;
  size_t bb = (size_t)s * 32 * N_ + n;
  for (int a = 0; a < KM_; a++) {
    float xr = exr[eb + (size_t)a * N_], xi = exi[eb + (size_t)a * N_];
    float ka = kvalf(a);
    size_t i0 = gb0 + (size_t)a * N_;
    size_t i1 = gb1 + (size_t)a * N_;
    size_t ib = bb + (size_t)a * N_;
    // part A: du/dx = -sum_a ka*Im(cEx*G); Im(cEx*G) = xr*Gi - xi*Gr
    gx += g0 * (-ka * (xr * GI[i0] - xi * GR[i0]));
    gy += g0 * (-(xr * GyI[i0] - xi * GyR[i0]));
    gz += g0 * (-(xr * GzI[i0] - xi * GzR[i0]));
    gx += g1 * (-ka * (xr * GI[i1] - xi * GR[i1]));
    gy += g1 * (-(xr * GyI[i1] - xi * GyR[i1]));
    gz += g1 * (-(xr * GzI[i1] - xi * GzR[i1]));
    // part B: dE/dx = sum_a ka*Im(Ex*GB); Im(Ex*GB) = xr*GBi + xi*GBr
    gx += ka * (xr * GBI[ib] + xi * GBR[ib]);
    gy += (xr * GByI[ib] + xi * GByR[ib]);
    gz += (xr * GBzI[ib] + xi * GBzR[ib]);
  }
  atomicAdd(&Fout[(size_t)t * 3 + 0], -gx * INVL3);
  atomicAdd(&Fout[(size_t)t * 3 + 1], -gy * INVL3);
  atomicAdd(&Fout[(size_t)t * 3 + 2], -gz * INVL3);
}

// ------------------------------ host side ----------------------------------

extern "C" void kernel_launch(void* const* d_in, const int* in_sizes, int n_in,
                              void* d_out, int out_size, void* d_ws,
                              size_t ws_size, hipStream_t stream) {
  // setup_inputs() dict flattening (lists expand in order):
  // 0:inputs 1:neigh 2-6:desc_W 7-11:desc_b 12-16:lr_W 17-21:lr_b
  // 22-26:fit_W 27-31:fit_b 32:lin_W 33:nufft_amp 34:nufft_mu
  const float* inp = (const float*)d_in[0];
  const int* neigh = (const int*)d_in[1];
  const float* dW[5]; const float* db_[5];
  const float* lW[5]; const float* lb[5];
  const float* fW[5]; const float* fb[5];
  for (int i = 0; i < 5; i++) {
    dW[i] = (const float*)d_in[2 + i];
    db_[i] = (const float*)d_in[7 + i];
    lW[i] = (const float*)d_in[12 + i];
    lb[i] = (const float*)d_in[17 + i];
    fW[i] = (const float*)d_in[22 + i];
    fb[i] = (const float*)d_in[27 + i];
  }
  const float* linW = (const float*)d_in[32];
  const float* amp = (const float*)d_in[33];
  const float* mu = (const float*)d_in[34];

  float* ws = (float*)d_ws;
  size_t off = 0;
  auto alloc = [&](size_t nf) { float* p = ws + off; off += nf; return p; };
  const size_t ESZ = (size_t)S_ * KM_ * N_;
  const size_t FSZ = (size_t)S_ * 32 * BCP_;
  const size_t HSZ = (size_t)S_ * C_ * 32 * BCP_;
  const size_t GSZ = (size_t)S_ * C_ * 32 * N_;
  const size_t BSZ = (size_t)S_ * 32 * N_;
  const size_t DSZ = (size_t)S_ * N_ * 32;
  const size_t USZ = (size_t)S_ * N_ * C_;
  float* exr = alloc(ESZ); float* exi = alloc(ESZ);
  float* eyr = alloc(ESZ); float* eyi = alloc(ESZ);
  float* ezr = alloc(ESZ); float* ezi = alloc(ESZ);
  float* fhatR = alloc(FSZ); float* fhatI = alloc(FSZ);
  float* hR = alloc(HSZ); float* hI = alloc(HSZ);
  float* WhR = alloc(HSZ); float* WhI = alloc(HSZ);
  float* W2R = alloc(FSZ); float* W2I = alloc(FSZ);
  float* GR = alloc(GSZ); float* GI = alloc(GSZ);
  float* GyR = alloc(GSZ); float* GyI = alloc(GSZ);
  float* GzR = alloc(GSZ); float* GzI = alloc(GSZ);
  float* GBR = alloc(BSZ); float* GBI = alloc(BSZ);
  float* GByR = alloc(BSZ); float* GByI = alloc(BSZ);
  float* GBzR = alloc(BSZ); float* GBzI = alloc(BSZ);
  float* Db = alloc(DSZ); float* gDb = alloc(DSZ);
  float* ub = alloc(USZ); float* gub = alloc(USZ);

  float* Eout = (float*)d_out;       // [S]
  float* Fout = Eout + S_;           // [S][N][3]

  // zero outputs (energy + force accumulators)
  k_zero<<<(out_size + 255) / 256, 256, 0, stream>>>((float*)d_out, out_size);

  // phases
  k_phase<<<(S_ * N_ + 255) / 256, 256, 0, stream>>>(inp, exr, exi, eyr, eyi,
                                                     ezr, ezi);
  // descriptor forward
  k_desc_fwd<<<S_ * N_, 64, 0, stream>>>(inp, neigh, dW[0], dW[1], dW[2], dW[3],
                                         dW[4], db_[0], db_[1], db_[2], db_[3],
                                         db_[4], Db);
  // forward transform f_hat (WMMA)
  k_gemm_fhat<false><<<dim3(61, 2, S_), 32, 0, stream>>>(
      exr, exi, eyr, eyi, ezr, ezi, nullptr, 1, 1.f, 1.f, fhatR, fhatI);
  // screened multiplier
  {
    size_t total = (size_t)S_ * C_ * 32 * BCP_;
    k_mult<<<(unsigned)((total + 255) / 256), 256, 0, stream>>>(fhatR, fhatI,
                                                                amp, mu, hR, hI);
  }
  // back-transforms G, Gy, Gz (WMMA)
  k_gemm_g<0><<<dim3(64, 2, S_ * C_), 32, 0, stream>>>(hR, hI, eyr, eyi, ezr,
                                                       ezi, C_, -1.f, GR, GI);
  k_gemm_g<1><<<dim3(64, 2, S_ * C_), 32, 0, stream>>>(hR, hI, eyr, eyi, ezr,
                                                       ezi, C_, -1.f, GyR, GyI);
  k_gemm_g<2><<<dim3(64, 2, S_ * C_), 32, 0, stream>>>(hR, hI, eyr, eyi, ezr,
                                                       ezi, C_, -1.f, GzR, GzI);
  // u
  k_u<<<(S_ * N_ + 255) / 256, 256, 0, stream>>>(exr, exi, GR, GI, ub);
  // fit + lr MLPs fwd/bwd, energy, adjoints
  k_mlp<<<(S_ * N_ + 63) / 64, 64, 0, stream>>>(
      Db, ub, lW[0], lW[1], lW[2], lW[3], lW[4], lb[0], lb[1], lb[2], lb[3],
      lb[4], fW[0], fW[1], fW[2], fW[3], fW[4], fb[0], fb[1], fb[2], fb[3],
      fb[4], linW, Eout, gDb, gub);
  // descriptor backward -> forces
  k_desc_bwd<<<S_ * N_, 64, 0, stream>>>(inp, neigh, dW[0], dW[1], dW[2], dW[3],
                                         dW[4], db_[0], db_[1], db_[2], db_[3],
                                         db_[4], gDb, Fout);
  // adjoint forward transform Wh (gu-weighted, conj E) (WMMA)
  k_gemm_fhat<true><<<dim3(61, 2, S_ * C_), 32, 0, stream>>>(
      exr, exi, eyr, eyi, ezr, ezi, gub, C_, -1.f, -1.f, WhR, WhI);
  // W2 = sum_ch mult * Wh
  {
    size_t total = (size_t)S_ * 32 * BCP_;
    k_w2<<<(unsigned)((total + 255) / 256), 256, 0, stream>>>(WhR, WhI, amp, mu,
                                                              W2R, W2I);
  }
  // adjoint back-transforms GB, GBy, GBz (WMMA, non-conj E)
  k_gemm_g<0><<<dim3(64, 2, S_), 32, 0, stream>>>(W2R, W2I, eyr, eyi, ezr, ezi,
                                                  1, 1.f, GBR, GBI);
  k_gemm_g<1><<<dim3(64, 2, S_), 32, 0, stream>>>(W2R, W2I, eyr, eyi, ezr, ezi,
                                                  1, 1.f, GByR, GByI);
  k_gemm_g<2><<<dim3(64, 2, S_), 32, 0, stream>>>(W2R, W2I, eyr, eyi, ezr, ezi,
                                                  1, 1.f, GBzR, GBzI);
  // long-range forces
  k_nufft_forces<<<(S_ * N_ + 255) / 256, 256, 0, stream>>>(
      exr, exi, GR, GI, GyR, GyI, GzR, GzI, GBR, GBI, GByR, GByI, GBzR, GBzI,
      gub, Fout);
  (void)in_sizes; (void)n_in; (void)ws_size;
}